// Prototype_8306466750631
// MI455X (gfx1250) — compile-verified
//
#include <hip/hip_runtime.h>
#include <hip/hip_bf16.h>

typedef __attribute__((ext_vector_type(2))) float v2f;
typedef __attribute__((ext_vector_type(8))) float v8f;

#define NCLASS 100
#define NB     512
#define H      768
#define MAXT   132            // >= NCLASS + NB/16 upper bound on row-tiles
#define LDSP   772            // LDS row stride (floats): 772 % 64 == 4 -> conflict-free 16 rows, 8B aligned

// ---------------------------------------------------------------------------
// Kernel 1: deterministic bucketing of batch rows by class into 16-row tiles.
// One block, 128 threads. tile_class[t] = class id (or -1 unused),
// tile_rows[t*16+r] = batch index (or -1 pad).
// ---------------------------------------------------------------------------
__global__ void bucket_kernel(const int* __restrict__ y,
                              int* __restrict__ tile_class,
                              int* __restrict__ tile_rows) {
  __shared__ int cnt[NCLASS];
  __shared__ int toff[NCLASS];
  const int tid = threadIdx.x;
  for (int i = tid; i < MAXT; i += blockDim.x)      tile_class[i] = -1;
  for (int i = tid; i < MAXT * 16; i += blockDim.x) tile_rows[i]  = -1;
  if (tid < NCLASS) {
    int c = 0;
    for (int b = 0; b < NB; ++b) c += (y[b] == tid) ? 1 : 0;
    cnt[tid] = c;
  }
  __syncthreads();
  if (tid == 0) {
    int off = 0;
    for (int c = 0; c < NCLASS; ++c) { toff[c] = off; off += (cnt[c] + 15) >> 4; }
  }
  __syncthreads();
  if (tid < NCLASS) {
    const int base   = toff[tid];
    const int ntile  = (cnt[tid] + 15) >> 4;
    for (int j = 0; j < ntile; ++j) tile_class[base + j] = tid;
    int idx = 0;
    for (int b = 0; b < NB; ++b) {
      if (y[b] == tid) {
        tile_rows[(base + (idx >> 4)) * 16 + (idx & 15)] = b;
        ++idx;
      }
    }
  }
}

// ---------------------------------------------------------------------------
// Kernel 2: T[t] = Xtile @ Mt[c]   (Mt = triu(M[c]), diag -> max(d, 0.001))
// grid = (6, MAXT), block = 256 (8 waves). Wave w owns output cols
// [bx*128 + 16w, +16). Triu => only h <= col contributes: k-loop ends at col0+16.
// Bulk region k0 < col0 is uniformly strict-upper-triangular -> no masking.
// Bulk loop: outer step 16, inner fully unrolled 4x (col0 is a multiple of 16)
// so 8 global loads clause together and one wait covers 4 WMMAs.
// ---------------------------------------------------------------------------
__global__ void gemm1_kernel(const float* __restrict__ x,
                             const float* __restrict__ M,
                             const int* __restrict__ tile_class,
                             const int* __restrict__ tile_rows,
                             float* __restrict__ T) {
  __shared__ float As[16 * LDSP];
  const int t = blockIdx.y;
  const int c = tile_class[t];
  if (c < 0) return;
  const int tid = threadIdx.x;

  // Stage gathered 16 x 768 A tile (padded rows -> 0)
  {
    const int r  = tid >> 4;
    const int lc = tid & 15;
    const int b  = tile_rows[t * 16 + r];
    const float* xr = x + (size_t)(b < 0 ? 0 : b) * H;
    for (int col = lc; col < H; col += 16)
      As[r * LDSP + col] = (b >= 0) ? xr[col] : 0.0f;
  }
  __syncthreads();

  const int wave = tid >> 5;
  const int lane = tid & 31;
  const int n    = lane & 15;    // B col / A row within tile
  const int hi   = lane >> 4;    // 0: K pair {0,1}, 1: K pair {2,3}
  const int col0 = blockIdx.x * 128 + wave * 16;
  const int colg = col0 + n;
  const float* Mc    = M + (size_t)c * H * H;
  const float* Mcol  = Mc + colg;            // walk down column colg
  const float* Asrow = &As[n * LDSP + hi * 2];

  v8f acc = {};

  // Bulk: h in [0, col0) -> strictly above-diagonal for all lanes, no mask.
  for (int k0 = 0; k0 < col0; k0 += 16) {
    #pragma unroll
    for (int u = 0; u < 16; u += 4) {
      const int kk = k0 + u;
      const v2f a = *(const v2f*)&Asrow[kk];
      const int h0 = kk + hi * 2;
      v2f bf;
      bf.x = Mcol[(size_t)h0 * H];
      bf.y = Mcol[(size_t)(h0 + 1) * H];
      acc = __builtin_amdgcn_wmma_f32_16x16x4_f32(false, a, false, bf,
                                                  (short)0, acc, false, false);
    }
  }
  // Diagonal band: h in [col0, col0+16) -> per-element triu mask + diag fix.
  #pragma unroll
  for (int u = 0; u < 16; u += 4) {
    const int kk = col0 + u;
    const v2f a = *(const v2f*)&Asrow[kk];
    const int h0 = kk + hi * 2;
    const float m0 = Mcol[(size_t)h0 * H];
    const float m1 = Mcol[(size_t)(h0 + 1) * H];
    v2f bf;
    bf.x = (colg > h0)     ? m0 : ((colg == h0)     ? fmaxf(m0, 0.001f) : 0.0f);
    bf.y = (colg > h0 + 1) ? m1 : ((colg == h0 + 1) ? fmaxf(m1, 0.001f) : 0.0f);
    acc = __builtin_amdgcn_wmma_f32_16x16x4_f32(false, a, false, bf,
                                                (short)0, acc, false, false);
  }

  float* Tt = T + (size_t)t * 16 * H;
  const int r0 = hi * 8;
  #pragma unroll
  for (int v = 0; v < 8; ++v)
    Tt[(size_t)(r0 + v) * H + colg] = acc[v];
}

// ---------------------------------------------------------------------------
// Kernel 3: out rows = T[t] @ Mt[c]^T ; B[k][j] = Mt[c][j][k] (nonzero k >= j)
// => k-loop starts at j0. Diagonal band [j0, j0+16) masked; bulk [j0+16, 768)
// is uniformly k > j for all lanes -> raw b64 loads, no masking. Bulk loop
// outer step 16 / inner unrolled 4x (768 - j0 - 16 is a multiple of 16).
// ---------------------------------------------------------------------------
__global__ void gemm2_kernel(const float* __restrict__ M,
                             const int* __restrict__ tile_class,
                             const int* __restrict__ tile_rows,
                             const float* __restrict__ T,
                             float* __restrict__ out) {
  __shared__ float As[16 * LDSP];
  const int t = blockIdx.y;
  const int c = tile_class[t];
  if (c < 0) return;
  const int tid = threadIdx.x;

  {
    const int r  = tid >> 4;
    const int lc = tid & 15;
    const float* Tt = T + (size_t)t * 16 * H;
    for (int col = lc; col < H; col += 16)
      As[r * LDSP + col] = Tt[(size_t)r * H + col];
  }
  __syncthreads();

  const int wave = tid >> 5;
  const int lane = tid & 31;
  const int n    = lane & 15;
  const int hi   = lane >> 4;
  const int j0   = blockIdx.x * 128 + wave * 16;
  const int jg   = j0 + n;                       // output col == Mt row
  const float* Mrow  = M + (size_t)c * H * H + (size_t)jg * H + hi * 2;
  const float* Asrow = &As[n * LDSP + hi * 2];

  v8f acc = {};

  // Diagonal band: k in [j0, j0+16) -> per-element triu mask + diag fix.
  #pragma unroll
  for (int u = 0; u < 16; u += 4) {
    const int kk = j0 + u;
    const v2f a  = *(const v2f*)&Asrow[kk];
    const int kx = kk + hi * 2;
    const v2f m  = *(const v2f*)&Mrow[kk];       // 8B-aligned consecutive pair
    v2f bf;
    bf.x = (kx > jg)     ? m.x : ((kx == jg)     ? fmaxf(m.x, 0.001f) : 0.0f);
    bf.y = (kx + 1 > jg) ? m.y : ((kx + 1 == jg) ? fmaxf(m.y, 0.001f) : 0.0f);
    acc = __builtin_amdgcn_wmma_f32_16x16x4_f32(false, a, false, bf,
                                                (short)0, acc, false, false);
  }
  // Bulk: k in [j0+16, 768) -> k > j for every lane, no mask.
  for (int k0 = j0 + 16; k0 < H; k0 += 16) {
    #pragma unroll
    for (int u = 0; u < 16; u += 4) {
      const int kk = k0 + u;
      const v2f a = *(const v2f*)&Asrow[kk];
      const v2f m = *(const v2f*)&Mrow[kk];
      acc = __builtin_amdgcn_wmma_f32_16x16x4_f32(false, a, false, m,
                                                  (short)0, acc, false, false);
    }
  }

  const int r0 = hi * 8;
  #pragma unroll
  for (int v = 0; v < 8; ++v) {
    const int b = tile_rows[t * 16 + r0 + v];
    if (b >= 0) out[(size_t)b * H + jg] = acc[v];
  }
}

// ---------------------------------------------------------------------------
extern "C" void kernel_launch(void* const* d_in, const int* in_sizes, int n_in,
                              void* d_out, int out_size, void* d_ws, size_t ws_size,
                              hipStream_t stream) {
  const float* x = (const float*)d_in[0];   // [512, 768]
  const int*   y = (const int*)d_in[1];     // [512]
  const float* M = (const float*)d_in[2];   // [100, 768, 768]
  float* out = (float*)d_out;               // [512, 768]

  int*   tile_class = (int*)d_ws;                       // 132 ints
  int*   tile_rows  = tile_class + MAXT;                // 132*16 ints
  float* T          = (float*)((char*)d_ws + 16384);    // 132*16*768 floats (~6.5 MB)

  bucket_kernel<<<1, 128, 0, stream>>>(y, tile_class, tile_rows);
  dim3 grid(H / 128, MAXT);   // (6, 132)
  gemm1_kernel<<<grid, 256, 0, stream>>>(x, M, tile_class, tile_rows, T);
  gemm2_kernel<<<grid, 256, 0, stream>>>(M, tile_class, tile_rows, T, out);
}